// LLaMAMoE_55551107006972
// MI455X (gfx1250) — compile-verified
//
#include <hip/hip_runtime.h>
#include <math.h>

// ---------------- problem constants (match reference) ----------------
#define Ndim 4096            // B*T tokens
#define Cdim 2048            // model dim
#define Idim 5632            // moe intermediate
#define Edim 8               // experts

// ---------------- tiling ----------------
#define TM   16              // tokens per workgroup tile
#define IC   256             // I-chunk (Idim / IC == 22 exactly)
#define NCHUNK (Idim / IC)
#define XS_STRIDE 2052       // 2048 + 4 pad: 16B-aligned rows (async B128), 4-bank row skew
#define HS_STRIDE 264        // 256 + 8 pad
#define WG   512             // 16 waves (wave32)

typedef __attribute__((ext_vector_type(2))) float v2f;
typedef __attribute__((ext_vector_type(8))) float v8f;

// ---------------------------------------------------------------
// Kernel 0: zero the output accumulator (harness poisons d_out).
// ---------------------------------------------------------------
__global__ void zero_f32(float* __restrict__ p, int n) {
    int i = blockIdx.x * blockDim.x + threadIdx.x;
    if (i < n) p[i] = 0.0f;
}

// ---------------------------------------------------------------
// Kernel 1: router.  comb[t][e] = softmax-over-top2 weight, else 0.
// ---------------------------------------------------------------
__global__ void router_kernel(const float* __restrict__ x,
                              const float* __restrict__ gate_w,
                              float* __restrict__ comb) {
    int t = blockIdx.x * blockDim.x + threadIdx.x;
    if (t >= Ndim) return;
    float acc[Edim];
    #pragma unroll
    for (int e = 0; e < Edim; ++e) acc[e] = 0.0f;
    const float* xr = x + (size_t)t * Cdim;
    for (int c = 0; c < Cdim; ++c) {
        float xv = xr[c];
        #pragma unroll
        for (int e = 0; e < Edim; ++e) acc[e] += xv * gate_w[e * Cdim + c];
    }
    int a0 = 0; float v0 = acc[0];
    #pragma unroll
    for (int e = 1; e < Edim; ++e) if (acc[e] > v0) { v0 = acc[e]; a0 = e; }
    int a1 = (a0 == 0) ? 1 : 0; float v1 = acc[a1];
    #pragma unroll
    for (int e = 0; e < Edim; ++e)
        if (e != a0 && acc[e] > v1) { v1 = acc[e]; a1 = e; }
    float p0 = 1.0f / (1.0f + __expf(v1 - v0));
    float p1 = 1.0f - p0;
    float* cr = comb + (size_t)t * Edim;
    #pragma unroll
    for (int e = 0; e < Edim; ++e) cr[e] = 0.0f;
    cr[a0] = p0;
    cr[a1] = p1;
}

// ---------------------------------------------------------------
// Kernel 2: fused MoE FFN for one (expert, 16-token tile).
//   x tile    : async-copied to LDS (GLOBAL_LOAD_ASYNC_TO_LDS_B128),
//               reused by all 16 waves across 22 I-chunks x {fc1,fc2}.
//   weights   : streamed global->VGPR->WMMA (each row consumed once by
//               one wave; LDS staging would be pure overhead).
//   phase A   : h[16,IC] = silu(x @ fc1_chunk^T) * (x @ fc2_chunk^T)
//   phase B   : y[16,2048] += h @ proj_chunk^T
//   phase C   : out += comb[:,e] * y   (f32 global atomics)
// WMMA f32 16x16x4 fragment layouts (wave32):
//   A(16x4):  lane = m + 16*(k/2), vgpr = k%2
//   B(4x16):  lane = n + 16*(k/2), vgpr = k%2
//   C/D:      lane = n + 16*(m/8), vgpr = m%8
// ---------------------------------------------------------------
__global__ __launch_bounds__(WG) void moe_ffn_kernel(
        const float* __restrict__ x,
        const float* __restrict__ fc1_w,
        const float* __restrict__ fc2_w,
        const float* __restrict__ proj_w,
        const float* __restrict__ comb,
        float* __restrict__ out) {
    const int tile = blockIdx.x;          // token tile
    const int e    = blockIdx.y;          // expert
    const int t0   = tile * TM;
    const int tid  = threadIdx.x;
    const int wave = tid >> 5;
    const int lane = tid & 31;
    const int ml   = lane & 15;           // m / n fragment index
    const int kh   = lane >> 4;           // k half-select (0/1)

    // ---- sparse skip: nothing in this tile routed to expert e -> bail.
    float any = 0.0f;
    #pragma unroll
    for (int m = 0; m < TM; ++m)
        any += __builtin_fabsf(comb[(size_t)(t0 + m) * Edim + e]);
    if (any == 0.0f) return;              // uniform across the workgroup

    extern __shared__ float smem[];
    float* xs = smem;                     // [TM][XS_STRIDE] f32 x tile
    float* hs = smem + TM * XS_STRIDE;    // [TM][HS_STRIDE] f32 h chunk

    // ---- async copy of the x tile into LDS (B128 = 4 floats / lane / op)
    // 16 rows x 2048 floats = 8192 x b128; 512 threads -> 16 ops each.
    {
        #pragma unroll
        for (int i = 0; i < (TM * Cdim) / (4 * WG); ++i) {
            int idx  = tid + i * WG;      // b128 index
            int row  = idx >> 9;          // 512 b128 per row
            int colf = (idx & 511) * 4;   // float column
            unsigned     dst = (unsigned)(uintptr_t)(xs + row * XS_STRIDE + colf);
            const float* src = x + (size_t)(t0 + row) * Cdim + colf;
            asm volatile("global_load_async_to_lds_b128 %0, %1, off"
                         :: "v"(dst), "v"(src) : "memory");
        }
        asm volatile("s_wait_asynccnt 0x0" ::: "memory");
    }

    const float* fc1e = fc1_w + (size_t)e * Idim * Cdim;
    const float* fc2e = fc2_w + (size_t)e * Idim * Cdim;
    const float* pje  = proj_w + (size_t)e * Cdim * Idim;

    v8f yacc[8];
    #pragma unroll
    for (int j = 0; j < 8; ++j) yacc[j] = (v8f){};

    for (int ch = 0; ch < NCHUNK; ++ch) {
        __syncthreads();                  // hs free to overwrite / xs ready

        // ================= phase A: h chunk =================
        {
            const int i0 = ch * IC + wave * 16;   // this wave's 16 I-rows
            const float* ap  = xs + ml * XS_STRIDE + 2 * kh;
            const float* b1p = fc1e + (size_t)(i0 + ml) * Cdim + 2 * kh;
            const float* b2p = fc2e + (size_t)(i0 + ml) * Cdim + 2 * kh;
            v8f acc1 = (v8f){};
            v8f acc2 = (v8f){};
            for (int c = 0; c < Cdim; c += 4) {
                v2f a  = *(const v2f*)(ap  + c);
                v2f b1 = *(const v2f*)(b1p + c);
                v2f b2 = *(const v2f*)(b2p + c);
                acc1 = __builtin_amdgcn_wmma_f32_16x16x4_f32(
                        false, a, false, b1, (short)0, acc1, false, false);
                acc2 = __builtin_amdgcn_wmma_f32_16x16x4_f32(
                        false, a, false, b2, (short)0, acc2, false, false);
            }
            // silu(acc1) * acc2 -> hs  (v_rcp_f32 instead of IEEE divide)
            #pragma unroll
            for (int r = 0; r < 8; ++r) {
                float g = acc1[r];
                float s = g * __builtin_amdgcn_rcpf(1.0f + __expf(-g));
                float h = s * acc2[r];
                int m = r + 8 * kh;
                hs[m * HS_STRIDE + wave * 16 + ml] = h;
            }
        }
        __syncthreads();                  // hs visible to all waves

        // ================= phase B: y += h @ proj_chunk^T =================
        {
            const int c0 = wave * 128;    // this wave's 128 output columns
            const float* apb = hs + ml * HS_STRIDE + 2 * kh;
            const size_t ibase = (size_t)ch * IC + 2 * kh;
            for (int k = 0; k < IC; k += 4) {
                v2f a = *(const v2f*)(apb + k);
                #pragma unroll
                for (int j = 0; j < 8; ++j) {
                    const float* bp = pje
                        + (size_t)(c0 + 16 * j + ml) * Idim + ibase + k;
                    v2f b = *(const v2f*)bp;
                    yacc[j] = __builtin_amdgcn_wmma_f32_16x16x4_f32(
                            false, a, false, b, (short)0, yacc[j], false, false);
                }
            }
        }
    }

    // ================= phase C: comb-scale + combine =================
    float cw[8];
    #pragma unroll
    for (int r = 0; r < 8; ++r)
        cw[r] = comb[(size_t)(t0 + r + 8 * kh) * Edim + e];
    const int c0 = wave * 128;
    #pragma unroll
    for (int j = 0; j < 8; ++j) {
        #pragma unroll
        for (int r = 0; r < 8; ++r) {
            int m = r + 8 * kh;
            atomicAdd(&out[(size_t)(t0 + m) * Cdim + c0 + 16 * j + ml],
                      yacc[j][r] * cw[r]);
        }
    }
}

// ---------------------------------------------------------------
// Host entry point
// ---------------------------------------------------------------
extern "C" void kernel_launch(void* const* d_in, const int* in_sizes, int n_in,
                              void* d_out, int out_size, void* d_ws, size_t ws_size,
                              hipStream_t stream) {
    (void)in_sizes; (void)n_in; (void)ws_size;
    const float* x      = (const float*)d_in[0];  // [N, C]
    const float* gate_w = (const float*)d_in[1];  // [E, C]
    const float* fc1_w  = (const float*)d_in[2];  // [E, I, C]
    const float* fc2_w  = (const float*)d_in[3];  // [E, I, C]
    const float* proj_w = (const float*)d_in[4];  // [E, C, I]
    float* out  = (float*)d_out;                  // [N, C]
    float* comb = (float*)d_ws;                   // [N, E] scratch

    zero_f32<<<(out_size + 255) / 256, 256, 0, stream>>>(out, out_size);
    router_kernel<<<Ndim / 256, 256, 0, stream>>>(x, gate_w, comb);

    size_t lds_bytes = (size_t)(TM * XS_STRIDE + TM * HS_STRIDE) * sizeof(float);
    dim3 grid(Ndim / TM, Edim);
    moe_ffn_kernel<<<grid, WG, lds_bytes, stream>>>(
        x, fc1_w, fc2_w, proj_w, comb, out);
}